// Wormologist_65867618452260
// MI455X (gfx1250) — compile-verified
//
#include <hip/hip_runtime.h>
#include <math.h>

typedef __attribute__((ext_vector_type(2))) float v2f;
typedef __attribute__((ext_vector_type(4))) float v4f;
typedef __attribute__((ext_vector_type(8))) float v8f;

#define HID   512
#define HALF  256
#define OUTD  558
#define NIN   64
#define LDH   516   // padded LDS stride for h  (516 mod 64 == 4 -> conflict-free row reads)
#define LDSW  260   // padded LDS stride for sw (260 mod 64 == 4)
#define WAVES 4     // waves per workgroup, all cooperating on one 16-row tile

// ---------------------------------------------------------------------------
// Zero-fill the output tensor (vectorized, non-temporal: it is write-once and
// should not evict W1/W2/x from the 192MB L2).
// ---------------------------------------------------------------------------
__global__ void Wormologist_zero_kernel(v4f* __restrict__ out, long long n4) {
    long long i = (long long)blockIdx.x * blockDim.x + threadIdx.x;
    if (i < n4) {
        v4f z = (v4f)(0.f);
        __builtin_nontemporal_store(z, &out[i]);
    }
}

__global__ void Wormologist_zero_tail_kernel(float* __restrict__ out,
                                             long long start, long long n) {
    long long i = start + (long long)blockIdx.x * blockDim.x + threadIdx.x;
    if (i < n) out[i] = 0.f;
}

// ---------------------------------------------------------------------------
// Fused node-MLP head: h = x@Wn+bn ; hh = h@W1+b1 ; sw = a*silu(g) ;
// logits = sw@W2+b2 ; scatter to out[batch, pos, :].
//
// One workgroup (4 waves, wave32) per 16-node tile. The 4 waves share the
// h / sw LDS tiles and split the N (output-column) dimension of every stage,
// giving 24 waves per WGP at ~50KB LDS per workgroup instead of 6.
// All matmuls via V_WMMA_F32_16X16X4_F32 (full fp32 precision).
// ---------------------------------------------------------------------------
__global__ __launch_bounds__(32 * WAVES) void Wormologist_gnn_head_kernel(
    const float* __restrict__ x,
    const float* __restrict__ W_node, const float* __restrict__ b_node,
    const float* __restrict__ W1,     const float* __restrict__ b1,
    const float* __restrict__ W2,     const float* __restrict__ b2,
    const int*   __restrict__ batch,  const int* __restrict__ pos,
    float* __restrict__ out, int N, int max_nodes)
{
    __shared__ float hs [16 * LDH];   // h  tile [16,512] (padded)
    __shared__ float sws[16 * LDSW];  // sw tile [16,256] (padded)
    __shared__ int   rowoff[16];

    const int lane = threadIdx.x & 31;   // lane within wave32
    const int wave = threadIdx.x >> 5;   // 0..3
    const int m    = lane & 15;          // matrix row (A) / column (B,C) index
    const int hi   = lane >> 4;          // half-wave select
    const int base = blockIdx.x * 16;
    const int node = base + m;
    const bool mvalid = node < N;

    // Precompute output row offsets for the scatter (first 16 threads).
    if (threadIdx.x < 16) {
        int idx = base + threadIdx.x;
        rowoff[threadIdx.x] =
            (idx < N) ? (batch[idx] * max_nodes + pos[idx]) * OUTD : -1;
    }

    // ---- Stage 1: h = x @ W_node + b_node    ([16,64] x [64,512]) ----------
    // A-fragment layout (16x4 f32): lanes 0-15 hold K = k0,k0+1; lanes 16-31
    // hold K = k0+2,k0+3 for row M = lane&15. Each wave keeps its own copy.
    v2f afrag[16];
#pragma unroll
    for (int kt = 0; kt < 16; ++kt) {
        const int k = kt * 4 + hi * 2;
        v2f a;
        a.x = mvalid ? x[node * NIN + k]     : 0.f;
        a.y = mvalid ? x[node * NIN + k + 1] : 0.f;
        afrag[kt] = a;
    }

    // 32 column tiles split across 4 waves (8 each).
    for (int nt = wave * 8; nt < wave * 8 + 8; ++nt) {
        const int n0 = nt * 16;
        const float bias = b_node[n0 + m];
        v8f c;
#pragma unroll
        for (int j = 0; j < 8; ++j) c[j] = bias;
#pragma unroll
        for (int kt = 0; kt < 16; ++kt) {
            const int k = kt * 4 + hi * 2;
            v2f b;                                    // B 4x16: row K, col = n0+m
            b.x = W_node[ k      * HID + n0 + m];
            b.y = W_node[(k + 1) * HID + n0 + m];
            c = __builtin_amdgcn_wmma_f32_16x16x4_f32(false, afrag[kt],
                                                      false, b, (short)0, c,
                                                      false, false);
        }
        // C layout: VGPR j holds rows j (lanes 0-15) and j+8 (lanes 16-31).
#pragma unroll
        for (int j = 0; j < 8; ++j)
            hs[(j + hi * 8) * LDH + n0 + m] = c[j];
    }
    __syncthreads();

    // ---- Stage 2: hh = h @ W1 + b1 ; sw = a * silu(g) ----------------------
    // Pair tile jt (cols of "a") with tile jt+16 (cols of "g"): identical
    // C-fragment coordinates -> SwiGLU is elementwise in registers.
    // 16 paired tiles split across 4 waves (4 each).
    for (int jt = wave * 4; jt < wave * 4 + 4; ++jt) {
        const int na = jt * 16;
        const int ng = HALF + jt * 16;
        const float biasA = b1[na + m];
        const float biasG = b1[ng + m];
        v8f ca, cg;
#pragma unroll
        for (int j = 0; j < 8; ++j) { ca[j] = biasA; cg[j] = biasG; }
#pragma unroll 8
        for (int kt = 0; kt < HID / 4; ++kt) {       // 128 K-steps
            const int k = kt * 4 + hi * 2;
            const v2f a = *(const v2f*)&hs[m * LDH + k];   // shared A fragment
            v2f bA, bG;
            bA.x = W1[ k      * HID + na + m];
            bA.y = W1[(k + 1) * HID + na + m];
            bG.x = W1[ k      * HID + ng + m];
            bG.y = W1[(k + 1) * HID + ng + m];
            ca = __builtin_amdgcn_wmma_f32_16x16x4_f32(false, a, false, bA,
                                                       (short)0, ca, false, false);
            cg = __builtin_amdgcn_wmma_f32_16x16x4_f32(false, a, false, bG,
                                                       (short)0, cg, false, false);
        }
#pragma unroll
        for (int j = 0; j < 8; ++j) {
            const float g  = cg[j];
            const float sw = ca[j] * (g / (1.f + expf(-g)));   // a * silu(g)
            sws[(j + hi * 8) * LDSW + na + m] = sw;
        }
    }
    __syncthreads();

    // ---- Stage 3: logits = sw @ W2 + b2 ; ragged scatter -------------------
    // 35 tiles (last partial) split across waves: 9,9,9,8.
    const int nt_beg = wave * 9;
    const int nt_end = (nt_beg + 9 < 35) ? nt_beg + 9 : 35;
    for (int nt = nt_beg; nt < nt_end; ++nt) {
        const int n0 = nt * 16;
        const int n  = n0 + m;
        const bool nvalid = n < OUTD;
        const float bias = nvalid ? b2[n] : 0.f;
        v8f c;
#pragma unroll
        for (int j = 0; j < 8; ++j) c[j] = bias;
#pragma unroll 8
        for (int kt = 0; kt < HALF / 4; ++kt) {      // 64 K-steps
            const int k = kt * 4 + hi * 2;
            const v2f a = *(const v2f*)&sws[m * LDSW + k];
            v2f b;
            b.x = nvalid ? W2[ k      * OUTD + n] : 0.f;
            b.y = nvalid ? W2[(k + 1) * OUTD + n] : 0.f;
            c = __builtin_amdgcn_wmma_f32_16x16x4_f32(false, a, false, b,
                                                      (short)0, c, false, false);
        }
#pragma unroll
        for (int j = 0; j < 8; ++j) {
            const int ro = rowoff[j + hi * 8];
            if (ro >= 0 && nvalid)
                __builtin_nontemporal_store(c[j], &out[ro + n]);
        }
    }
}

// ---------------------------------------------------------------------------
// Inputs (setup_inputs order):
//  0 x [N,64]  1 edge_attr  2 W_node[64,512]  3 b_node[512]  4 W_edge  5 b_edge
//  6 W1[512,512]  7 b1[512]  8 W2[256,558]  9 b2[558]  10 edge_index
// 11 batch[N]  12 pos[N]  13 max_nodes (scalar)
// Edge encoder output is dead in the reference -> skipped entirely.
// ---------------------------------------------------------------------------
extern "C" void kernel_launch(void* const* d_in, const int* in_sizes, int n_in,
                              void* d_out, int out_size, void* d_ws, size_t ws_size,
                              hipStream_t stream) {
    (void)n_in; (void)d_ws; (void)ws_size;
    const float* x      = (const float*)d_in[0];
    const float* W_node = (const float*)d_in[2];
    const float* b_node = (const float*)d_in[3];
    const float* W1     = (const float*)d_in[6];
    const float* b1     = (const float*)d_in[7];
    const float* W2     = (const float*)d_in[8];
    const float* b2     = (const float*)d_in[9];
    const int*   batch  = (const int*)d_in[11];
    const int*   pos    = (const int*)d_in[12];
    float*       out    = (float*)d_out;

    const int N         = in_sizes[0] / NIN;
    const int max_nodes = out_size / (64 * OUTD);   // out = [64, max_nodes, 558]

    // 1) zero-fill output (poisoned by harness; scatter only touches valid rows)
    const long long total = (long long)out_size;
    const long long n4    = total >> 2;
    if (n4 > 0) {
        const int zt = 256;
        const long long zb = (n4 + zt - 1) / zt;
        Wormologist_zero_kernel<<<dim3((unsigned)zb), zt, 0, stream>>>(
            (v4f*)d_out, n4);
    }
    const long long rem = total - (n4 << 2);
    if (rem > 0) {
        Wormologist_zero_tail_kernel<<<1, 64, 0, stream>>>(out, n4 << 2, total);
    }

    // 2) fused MLP head + scatter: one workgroup (4 cooperating waves) per
    //    16-node tile.
    const int blocks = (N + 15) / 16;
    Wormologist_gnn_head_kernel<<<blocks, 32 * WAVES, 0, stream>>>(
        x, W_node, b_node, W1, b1, W2, b2, batch, pos, out, N, max_nodes);
}